// GINEConvGraphGymLayer_13048110645793
// MI455X (gfx1250) — compile-verified
//
#include <hip/hip_runtime.h>
#include <hip/hip_bf16.h>

// ---------------------------------------------------------------------------
// GINEConv layer for MI455X (gfx1250, wave32).
//   msg  = relu(x[src] + edge_attr)            (bandwidth + L2 f32 atomics)
//   agg  = segment_sum(msg, dst) ; h = x + agg (EPS = 0)
//   h1   = relu(h @ W1 + b1) ; out = h1 @ W2 + b2   (fp32 WMMA 16x16x4)
// ---------------------------------------------------------------------------

typedef float v8f __attribute__((ext_vector_type(8)));
typedef float v2f __attribute__((ext_vector_type(2)));

#define D 128            // feature dim (both in and out)
#define LDS_PITCH 132    // 128 + 4 pad: 16B-aligned rows, conflict-free ds reads

// ---------------------------------------------------------------- init: agg = x
__global__ __launch_bounds__(256) void init_agg_kernel(const float* __restrict__ x,
                                                       float* __restrict__ agg,
                                                       int n_vec4) {
  int i = blockIdx.x * blockDim.x + threadIdx.x;
  if (i < n_vec4) {
    ((float4*)agg)[i] = ((const float4*)x)[i];
  }
}

// ------------------------------------------------- per-edge message + scatter
// One wave per edge: 32 lanes x float4 = 128 features. relu(x_src + e) is
// accumulated into agg[dst] with global_atomic_add_f32 (no return -> STOREcnt,
// RMW resolves in L2 since agg (51MB) fits the 192MB L2).
__global__ __launch_bounds__(256) void edge_scatter_kernel(
    const float* __restrict__ x, const float* __restrict__ eattr,
    const int* __restrict__ src, const int* __restrict__ dst,
    float* __restrict__ agg, int n_edges) {
  int e = blockIdx.x * 8 + (int)(threadIdx.x >> 5);
  if (e >= n_edges) return;
  int lane = threadIdx.x & 31;
  int s = src[e];
  int d = dst[e];
  float4 xv = ((const float4*)(x + (size_t)s * D))[lane];
  float4 ev = ((const float4*)(eattr + (size_t)e * D))[lane];
  float4 m;
  m.x = fmaxf(xv.x + ev.x, 0.0f);
  m.y = fmaxf(xv.y + ev.y, 0.0f);
  m.z = fmaxf(xv.z + ev.z, 0.0f);
  m.w = fmaxf(xv.w + ev.w, 0.0f);
  float* ap = agg + (size_t)d * D + lane * 4;
  atomicAdd(ap + 0, m.x);
  atomicAdd(ap + 1, m.y);
  atomicAdd(ap + 2, m.z);
  atomicAdd(ap + 3, m.w);
}

// ----------------------------------------------------- fp32 WMMA GEMM layer
// Y[N,128] = act(X[N,128] @ W[128,128] + b)   with act = relu or identity.
// 256 threads = 8 waves. Wave w owns output columns [16w, 16w+16).
// Each wave preloads its 128x16 W slice into 64 VGPRs (v2f bfrag[32]) once,
// then grid-strides over 16-row tiles: A tile staged in LDS, 32 chained
// v_wmma_f32_16x16x4_f32 accumulate the 16x16 fp32 output tile.
//
// V_WMMA_F32_16X16X4_F32 operand layout (wave32):
//   A 16x4:  lane l holds row M=l&15; VGPR0 = K=koff, VGPR1 = K=koff+1,
//            koff = 2*(l>>4)            (K = 0,1 lanes 0-15; K = 2,3 lanes 16-31)
//   B 4x16:  lane l holds col N=l&15; same K striping as A.
//   C/D:     VGPR r, lane l -> element (M = r + 8*(l>>4), N = l&15).
template <bool RELU>
__global__ __launch_bounds__(256) void mlp_layer_kernel(
    const float* __restrict__ X, const float* __restrict__ W,
    const float* __restrict__ bias, float* __restrict__ Y,
    int n_rows, int n_tiles) {
  __shared__ float sA[16 * LDS_PITCH];

  const int lane = threadIdx.x & 31;
  const int wave = threadIdx.x >> 5;          // 0..7 -> column tile
  const int ncol = wave * 16 + (lane & 15);   // this lane's output column
  const int koff = (lane >> 4) * 2;           // 0 or 2 (K striping)
  const int arow = lane & 15;                 // this lane's A row within tile

  // Preload B fragments: W[k][ncol] for all k, striped per WMMA layout.
  v2f bfrag[32];
#pragma unroll
  for (int kk = 0; kk < 32; ++kk) {
    int k = kk * 4 + koff;
    bfrag[kk].x = W[(size_t)k * D + ncol];
    bfrag[kk].y = W[(size_t)(k + 1) * D + ncol];
  }
  const float bv = bias[ncol];

  for (int tile = blockIdx.x; tile < n_tiles; tile += gridDim.x) {
    const int row0 = tile * 16;

    __syncthreads();  // previous tile's sA reads complete before overwrite
    // Stage A tile (16x128) into LDS: 512 float4 / 256 threads = 2 each.
#pragma unroll
    for (int i = threadIdx.x; i < 512; i += 256) {
      int r = i >> 5;   // row 0..15
      int c = i & 31;   // float4 column 0..31
      float4 v = ((const float4*)(X + (size_t)(row0 + r) * D))[c];
      *(float4*)&sA[r * LDS_PITCH + c * 4] = v;
    }
    __syncthreads();

    v8f acc = {};
#pragma unroll
    for (int kk = 0; kk < 32; ++kk) {
      int k = kk * 4 + koff;
      v2f afrag;
      afrag.x = sA[arow * LDS_PITCH + k];
      afrag.y = sA[arow * LDS_PITCH + k + 1];
      acc = __builtin_amdgcn_wmma_f32_16x16x4_f32(
          /*neg_a=*/false, afrag, /*neg_b=*/false, bfrag[kk],
          /*c_mod=*/(short)0, acc, /*reuse_a=*/false, /*reuse_b=*/false);
    }

    // Fused bias (+relu) epilogue and store.
#pragma unroll
    for (int r = 0; r < 8; ++r) {
      int m = r + 8 * (lane >> 4);
      float v = acc[r] + bv;
      if (RELU) v = fmaxf(v, 0.0f);
      int grow = row0 + m;
      if (grow < n_rows) Y[(size_t)grow * D + ncol] = v;
    }
  }
}

// ---------------------------------------------------------------------------
extern "C" void kernel_launch(void* const* d_in, const int* in_sizes, int n_in,
                              void* d_out, int out_size, void* d_ws, size_t ws_size,
                              hipStream_t stream) {
  const float* x     = (const float*)d_in[0];  // [N,128]
  const float* eattr = (const float*)d_in[1];  // [E,128]
  const float* W1    = (const float*)d_in[2];  // [128,128]
  const float* b1    = (const float*)d_in[3];  // [128]
  const float* W2    = (const float*)d_in[4];  // [128,128]
  const float* b2    = (const float*)d_in[5];  // [128]
  const int*   ei    = (const int*)d_in[6];    // [2,E] flattened

  const int n_nodes = in_sizes[0] / D;
  const int n_edges = in_sizes[1] / D;
  const int* src = ei;            // edge_index[0]
  const int* dst = ei + n_edges;  // edge_index[1]

  float* agg = (float*)d_ws;                          // [N,128] = h = x + agg
  float* h1  = agg + (size_t)n_nodes * D;             // [N,128] after layer 1
  float* out = (float*)d_out;                         // [N,128]

  // 1) agg = x   (EPS = 0 -> (1+EPS)*x folded into the accumulator init)
  {
    int n_vec4 = n_nodes * (D / 4);
    int grid = (n_vec4 + 255) / 256;
    init_agg_kernel<<<grid, 256, 0, stream>>>(x, agg, n_vec4);
  }

  // 2) scatter relu(x[src] + e) into agg[dst]   (8 edges per 256-thread block)
  {
    int grid = (n_edges + 7) / 8;
    edge_scatter_kernel<<<grid, 256, 0, stream>>>(x, eattr, src, dst, agg, n_edges);
  }

  // 3) h1 = relu(agg @ W1 + b1) ; 4) out = h1 @ W2 + b2
  {
    int n_tiles = (n_nodes + 15) / 16;
    int grid = n_tiles < 1536 ? n_tiles : 1536;  // persistent blocks reuse W regs
    mlp_layer_kernel<true><<<grid, 256, 0, stream>>>(agg, W1, b1, h1, n_nodes, n_tiles);
    mlp_layer_kernel<false><<<grid, 256, 0, stream>>>(h1, W2, b2, out, n_nodes, n_tiles);
  }
}